// GraphGather_29746943492201
// MI455X (gfx1250) — compile-verified
//
#include <hip/hip_runtime.h>
#include <cstdint>
#include <cmath>

// Problem constants (match reference): D_FEAT=256, out row stride = 512 (sum | max).
#define D_FEAT        256
#define OUT_STRIDE    512
#define ROWS_PER_BLK  512
#define NTHREADS      256

typedef float v4f __attribute__((ext_vector_type(4)));

// ---------------- CDNA5-specific primitives (inline asm) ----------------

// No-return device-scope f32 atomic add:  GLOBAL_ATOMIC_ADD_F32
__device__ __forceinline__ void atomic_add_f32(float* p, float v) {
  asm volatile("global_atomic_add_f32 %0, %1, off scope:SCOPE_DEV"
               :: "v"(p), "v"(v) : "memory");
}

// No-return device-scope f32 atomic max:  GLOBAL_ATOMIC_MAX_NUM_F32 (CDNA5 _num_ op)
__device__ __forceinline__ void atomic_max_f32(float* p, float v) {
  asm volatile("global_atomic_max_num_f32 %0, %1, off scope:SCOPE_DEV"
               :: "v"(p), "v"(v) : "memory");
}

// Async global->LDS dword copy (tracked by ASYNCcnt):  GLOBAL_LOAD_ASYNC_TO_LDS_B32
__device__ __forceinline__ void async_load_dw_to_lds(uint32_t lds_off, const int* gaddr) {
  asm volatile("global_load_async_to_lds_b32 %0, %1, off"
               :: "v"(lds_off), "v"(gaddr) : "memory");
}

__device__ __forceinline__ void wait_asynccnt0() {
  asm volatile("s_wait_asynccnt 0x0" ::: "memory");
}

// ---------------- kernels ----------------

__global__ void seg_init_kernel(float* __restrict__ out, int total) {
  int i = blockIdx.x * blockDim.x + threadIdx.x;
  if (i < total) {
    int c = i & (OUT_STRIDE - 1);
    out[i] = (c < D_FEAT) ? 0.0f : -__builtin_inff();
  }
}

__device__ __forceinline__ void flush_seg(float* __restrict__ out, int seg, int col,
                                          const v4f& s, const v4f& mx) {
  float* ps = out + (size_t)seg * OUT_STRIDE + col;
  atomic_add_f32(ps + 0, s.x);
  atomic_add_f32(ps + 1, s.y);
  atomic_add_f32(ps + 2, s.z);
  atomic_add_f32(ps + 3, s.w);
  float* pm = ps + D_FEAT;  // max half lives at columns [256, 512)
  atomic_max_f32(pm + 0, mx.x);
  atomic_max_f32(pm + 1, mx.y);
  atomic_max_f32(pm + 2, mx.z);
  atomic_max_f32(pm + 3, mx.w);
}

__global__ __launch_bounds__(NTHREADS) void seg_reduce_kernel(
    const float* __restrict__ feat, const int* __restrict__ mem,
    float* __restrict__ out, int n_atoms) {
  __shared__ int s_m[ROWS_PER_BLK];

  const int tid  = threadIdx.x;
  const int base = blockIdx.x * ROWS_PER_BLK;
  const int rows = min(ROWS_PER_BLK, n_atoms - base);
  if (rows <= 0) return;  // uniform per block: safe w.r.t. barriers

  // Stage this chunk's (sorted) membership ids into LDS via async-to-LDS DMA.
  #pragma unroll
  for (int i = tid; i < ROWS_PER_BLK; i += NTHREADS) {
    int src = base + i;
    if (src >= n_atoms) src = n_atoms - 1;  // clamp (never consumed past `rows`)
    async_load_dw_to_lds((uint32_t)(uintptr_t)&s_m[i], mem + src);
  }
  wait_asynccnt0();
  __syncthreads();

  // Thread t owns columns [4*(t&63), 4*(t&63)+3] of rows {t>>6, (t>>6)+4, ...}.
  // Each wave32 issues contiguous 512B b128 loads; (row, segment id, boundary
  // branch) are wave-uniform -> force SGPR compare + s_cbranch via readfirstlane.
  const int rlane = tid >> 6;         // 0..3 row substream (constant per wave)
  const int col   = (tid & 63) << 2;  // column base, step 4
  const float* fp = feat + (size_t)base * D_FEAT + col;

  v4f s  = (v4f)(0.f);
  v4f mx = (v4f)(0.f);
  int cur = -1;

  #pragma unroll 2
  for (int r = rlane; r < rows; r += 4) {
    // Streamed exactly once: non-temporal so the 1GB stream doesn't evict the
    // atomically-updated 8MB accumulator region from the 192MB L2.
    const v4f v = __builtin_nontemporal_load((const v4f*)(fp + (size_t)r * D_FEAT));
    const int m = __builtin_amdgcn_readfirstlane(s_m[r]);  // wave-uniform -> SGPR
    if (m != cur) {                 // scalar branch (s_cmp/s_cbranch)
      if (cur >= 0) flush_seg(out, cur, col, s, mx);
      cur = m;
      s = v;                         // start new segment with this row
      mx = v;
    } else {
      s += v;
      mx.x = fmaxf(mx.x, v.x); mx.y = fmaxf(mx.y, v.y);
      mx.z = fmaxf(mx.z, v.z); mx.w = fmaxf(mx.w, v.w);
    }
  }
  if (cur >= 0) flush_seg(out, cur, col, s, mx);
}

__global__ void seg_tanh_kernel(float* __restrict__ out, int total) {
  int i = blockIdx.x * blockDim.x + threadIdx.x;
  if (i < total) out[i] = tanhf(out[i]);  // tanh(-inf) = -1 matches empty-segment max
}

// ---------------- host entry ----------------

extern "C" void kernel_launch(void* const* d_in, const int* in_sizes, int n_in,
                              void* d_out, int out_size, void* d_ws, size_t ws_size,
                              hipStream_t stream) {
  const float* feat = (const float*)d_in[0];
  const int*   mem  = (const int*)d_in[1];
  float*       out  = (float*)d_out;
  const int n_atoms = in_sizes[1];          // 1048576
  const int total   = out_size;             // 4096 * 512

  seg_init_kernel<<<(total + 255) / 256, 256, 0, stream>>>(out, total);

  const int nblocks = (n_atoms + ROWS_PER_BLK - 1) / ROWS_PER_BLK;
  seg_reduce_kernel<<<nblocks, NTHREADS, 0, stream>>>(feat, mem, out, n_atoms);

  seg_tanh_kernel<<<(total + 255) / 256, 256, 0, stream>>>(out, total);
}